// wideResNet_16415365005539
// MI455X (gfx1250) — compile-verified
//
#include <hip/hip_runtime.h>
#include <stdint.h>

// ---------------------------------------------------------------------------
// CDNA5 / gfx1250 implementation of the L_inf norm-distance-conv IBP ResNet.
// wave32; async-to-LDS staging of zero-padded input tiles (ASYNCcnt path);
// channel-mean normalization via chained V_WMMA_F32_16X16X4_F32.
// Kernel templated on (kernel size K, stride S, tile cols C) so the inner
// max-abs loops fully unroll and address math folds to constants.
// ---------------------------------------------------------------------------

typedef __attribute__((ext_vector_type(2))) float v2f;
typedef __attribute__((ext_vector_type(8))) float v8f;

#define TPOS  16    // output positions per workgroup tile
#define CMAX  128   // max output channels
#define CCH   16    // input-channel chunk staged in LDS
#define TILEE 100   // max per-channel footprint: 3x33 (k3 s2 C16)
#define NB    4     // batch

struct ConvArgs {
  const float* x;  const float* lo; const float* up;   // input streams (B,cin,Hin,Win)
  const float* w;                                      // (cout, cin*k*k)
  float* outx; float* outl; float* outu;               // outputs (B,cout,Hout,Wout)
  const float* bx; const float* bl; const float* bu;   // residual blend source (or null)
  int cin, cout;
  int Hin, Win, Hout, Wout;
  float alpha, beta;                                   // out = alpha*norm + beta*blend
};

template <int K, int S, int C>
__global__ __launch_bounds__(128) void ndconv_wmma_kernel(ConvArgs a) {
  constexpr int PAD  = (K == 3) ? 1 : 0;
  constexpr int R    = TPOS / C;              // output rows per tile
  constexpr int Rin  = (R - 1) * S + K;       // input rows in footprint
  constexpr int Cin_ = (C - 1) * S + K;       // input cols in footprint
  constexpr int nElem = Rin * Cin_;
  constexpr int KK   = K * K;

  __shared__ float tile[3][CCH][TILEE]; // zero-padded input footprint (x, lo, up)
  __shared__ float sd[CMAX][TPOS];      // running max |x - w|
  __shared__ float slo[CMAX][TPOS];     // running lower-bound distance
  __shared__ float sup[CMAX][TPOS];     // running upper-bound distance
  __shared__ float smean[3][TPOS];      // per-position channel means

  const int tid   = threadIdx.x;
  const int lane  = tid & 31;
  const int wave  = tid >> 5;
  const int n     = lane & 15;                 // position within the 16-pos tile
  const int slice = (wave << 1) | (lane >> 4); // 0..7 channel slice

  const int HW          = a.Hout * a.Wout;
  const int tilesPerImg = HW / TPOS;           // all sizes are multiples of 16
  const int b      = blockIdx.x / tilesPerImg;
  const int tileId = blockIdx.x - b * tilesPerImg;

  const int p0   = tileId * TPOS;
  const int y0   = p0 / a.Wout;
  const int x0   = p0 - y0 * a.Wout;
  const int iy0  = y0 * S - PAD;
  const int ix0  = x0 * S - PAD;
  const int py   = n / C;
  const int px   = n - py * C;
  const int p    = p0 + n;                     // linear output position

  // ---- channel-chunked accumulation with async-to-LDS staging --------------
  for (int cb = 0; cb < a.cin; cb += CCH) {
    const int cc = (a.cin - cb < CCH) ? (a.cin - cb) : CCH;
    __syncthreads();                           // previous chunk's compute done

    // Stage zero-padded tile: interior via GLOBAL_LOAD_ASYNC_TO_LDS_B32,
    // out-of-bounds border elements via plain DS stores of 0.
    for (int s = 0; s < 3; ++s) {
      const float* base = (s == 0) ? a.x : (s == 1) ? a.lo : a.up;
      const int lim = cc * nElem;
      for (int e = tid; e < lim; e += 128) {
        const int c  = e / nElem;              // compile-time divisors
        const int rc = e - c * nElem;
        const int r  = rc / Cin_;
        const int cl = rc - r * Cin_;
        const int iy = iy0 + r;
        const int ix = ix0 + cl;
        float* dst = &tile[s][c][rc];
        if ((unsigned)iy < (unsigned)a.Hin && (unsigned)ix < (unsigned)a.Win) {
          const float* g = base + (((size_t)b * a.cin + (cb + c)) * a.Hin + iy) * a.Win + ix;
          const uint32_t laddr = (uint32_t)(uintptr_t)dst;
          const uint64_t gaddr = (uint64_t)(uintptr_t)g;
          asm volatile("global_load_async_to_lds_b32 %0, %1, off"
                       :: "v"(laddr), "v"(gaddr) : "memory");
        } else {
          *dst = 0.0f;                         // zero padding (matches patches)
        }
      }
    }
    asm volatile("s_wait_asynccnt 0x0" ::: "memory");
    __syncthreads();

    // Branch-free, fully-unrolled inner loop over this chunk's K elements.
    for (int o = slice; o < a.cout; o += 8) {
      const float* wrow = a.w + (size_t)o * (a.cin * KK) + (size_t)cb * KK;
      __builtin_prefetch(wrow, 0, 1);          // -> global_prefetch_b8
      float d = 0.f, dl = 0.f, du = 0.f;
      for (int c = 0; c < cc; ++c) {
        const float* tx = &tile[0][c][0];
        const float* tl = &tile[1][c][0];
        const float* tu = &tile[2][c][0];
        const float* wc = wrow + c * KK;
#pragma unroll
        for (int dy = 0; dy < K; ++dy) {
          const int rbase = (py * S + dy) * Cin_ + px * S;
#pragma unroll
          for (int dx = 0; dx < K; ++dx) {
            const float wv = wc[dy * K + dx];
            const float xv = tx[rbase + dx];
            const float lv = tl[rbase + dx];
            const float uv = tu[rbase + dx];
            d  = fmaxf(d,  fabsf(xv - wv));
            dl = fmaxf(dl, fmaxf(fmaxf(lv - wv, wv - uv), 0.f));
            du = fmaxf(du, fmaxf(fabsf(lv - wv), fabsf(uv - wv)));
          }
        }
      }
      if (cb == 0) {
        sd[o][n] = d; slo[o][n] = dl; sup[o][n] = du;
      } else {                                 // each (o,n) owned by one thread
        sd[o][n]  = fmaxf(sd[o][n],  d);
        slo[o][n] = fmaxf(slo[o][n], dl);
        sup[o][n] = fmaxf(sup[o][n], du);
      }
    }
  }
  __syncthreads();

  // ---- channel mean per position via chained V_WMMA_F32_16X16X4_F32 --------
  // A = uniform 1/cout (16x4). D[m,n] = (1/cout) * sum_k B[k,n]: every D row
  // holds the per-position means, so lane L reads mean for N = L%16 from d[0].
  if (wave == 0) {
    const float inv = 1.0f / (float)a.cout;
    v2f av; av[0] = inv; av[1] = inv;
    v8f cd = {}, cl = {}, cu = {};
    const int half2 = (lane >> 4) << 1;
    for (int kb = 0; kb < a.cout; kb += 4) {
      const int k0 = kb + half2;
      v2f bd, bl, bu;
      bd[0] = sd[k0][n];  bd[1] = sd[k0 + 1][n];
      bl[0] = slo[k0][n]; bl[1] = slo[k0 + 1][n];
      bu[0] = sup[k0][n]; bu[1] = sup[k0 + 1][n];
      cd = __builtin_amdgcn_wmma_f32_16x16x4_f32(false, av, false, bd, (short)0, cd, false, false);
      cl = __builtin_amdgcn_wmma_f32_16x16x4_f32(false, av, false, bl, (short)0, cl, false, false);
      cu = __builtin_amdgcn_wmma_f32_16x16x4_f32(false, av, false, bu, (short)0, cu, false, false);
    }
    if (lane < 16) {
      smean[0][n] = cd[0];   // mean(d)
      smean[1][n] = cl[0];   // mean(dl)
      smean[2][n] = cu[0];   // mean(du)
    }
  }
  __syncthreads();

  // ---- normalize (+ optional residual blend) and store ---------------------
  const float md  = smean[0][n];
  const float mdl = smean[1][n];
  const float mdu = smean[2][n];
  for (int o = slice; o < a.cout; o += 8) {
    const size_t oidx = ((size_t)b * a.cout + o) * HW + p;
    float vo = sd[o][n]  - md;    // out - mean(out)
    float vl = slo[o][n] - mdu;   // dl - mean(du)
    float vu = sup[o][n] - mdl;   // du - mean(dl)
    if (a.bx) {
      vo = a.alpha * vo + a.beta * a.bx[oidx];
      vl = a.alpha * vl + a.beta * a.bl[oidx];
      vu = a.alpha * vu + a.beta * a.bu[oidx];
    }
    a.outx[oidx] = vo;
    a.outl[oidx] = vl;
    a.outu[oidx] = vu;
  }
}

// Final global average pool: out[z,b,c] = mean_{h,w} stream_z[b,c,h,w]
__global__ void gap_kernel(const float* sx, const float* sl, const float* su,
                           float* out, int BC, int HW) {
  const int idx = blockIdx.x * blockDim.x + threadIdx.x;
  if (idx >= 3 * BC) return;
  const int z  = idx / BC;
  const int rc = idx - z * BC;
  const float* src = (z == 0 ? sx : (z == 1 ? sl : su)) + (size_t)rc * HW;
  float s = 0.f;
  for (int i = 0; i < HW; ++i) s += src[i];
  out[idx] = s / (float)HW;
}

// ---------------------------------------------------------------------------
static void launch_ndconv(const float* x, const float* lo, const float* up,
                          const float* w, float* ox, float* ol, float* ou,
                          const float* bx, const float* bl, const float* bu,
                          int cin, int cout, int k, int stride, int pad,
                          int Hin, int Win, float alpha, float beta,
                          hipStream_t stream) {
  ConvArgs a;
  a.x = x; a.lo = lo; a.up = up; a.w = w;
  a.outx = ox; a.outl = ol; a.outu = ou;
  a.bx = bx; a.bl = bl; a.bu = bu;
  a.cin = cin; a.cout = cout;
  a.Hin = Hin; a.Win = Win;
  a.Hout = (Hin + 2 * pad - k) / stride + 1;
  a.Wout = (Win + 2 * pad - k) / stride + 1;
  a.alpha = alpha; a.beta = beta;
  const int grid = NB * (a.Hout * a.Wout / TPOS);
  const int C = (a.Wout < TPOS) ? a.Wout : TPOS;

#define LAUNCH_ND(KK_, SS_, CC_) \
  ndconv_wmma_kernel<KK_, SS_, CC_><<<grid, 128, 0, stream>>>(a)

  if (k == 3) {
    if (stride == 1) { if (C == 16) LAUNCH_ND(3, 1, 16); else LAUNCH_ND(3, 1, 8); }
    else             { if (C == 16) LAUNCH_ND(3, 2, 16); else LAUNCH_ND(3, 2, 8); }
  } else { // k == 1
    if (stride == 1) { LAUNCH_ND(1, 1, 16); }
    else             { if (C == 16) LAUNCH_ND(1, 2, 16); else LAUNCH_ND(1, 2, 8); }
  }
#undef LAUNCH_ND
}

extern "C" void kernel_launch(void* const* d_in, const int* in_sizes, int n_in,
                              void* d_out, int out_size, void* d_ws, size_t ws_size,
                              hipStream_t stream) {
  (void)out_size; (void)ws_size;
  const float* X  = (const float*)d_in[0];
  const float* LO = (const float*)d_in[1];
  const float* UP = (const float*)d_in[2];

  const float* Wstem = nullptr;
  const float* Wc1[3][4] = {};
  const float* Wc2[3][4] = {};
  const float* Wsh[3] = {};
  static const int CH[4] = {16, 32, 64, 128};

  if (n_in > 4) {
    // One device pointer per pytree leaf: route by element count. Within each
    // repeated-size bucket, rely only on block order + conv1-before-conv2.
    int seq[3] = {0, 0, 0};
    for (int i = 3; i < n_in; ++i) {
      const float* ptr = (const float*)d_in[i];
      const int s = in_sizes[i];
      int g = -1;
      switch (s) {
        case 432:    Wstem     = ptr; break;  // 16 x 27
        case 4608:   Wc1[0][0] = ptr; break;  // 32 x 144
        case 512:    Wsh[0]    = ptr; break;  // 32 x 16
        case 18432:  Wc1[1][0] = ptr; break;  // 64 x 288
        case 2048:   Wsh[1]    = ptr; break;  // 64 x 32
        case 73728:  Wc1[2][0] = ptr; break;  // 128 x 576
        case 8192:   Wsh[2]    = ptr; break;  // 128 x 64
        case 9216:   g = 0; break;            // 32 x 288   (7 leaves)
        case 36864:  g = 1; break;            // 64 x 576   (7 leaves)
        case 147456: g = 2; break;            // 128 x 1152 (7 leaves)
        default: break;
      }
      if (g >= 0) {
        const int t = seq[g]++;
        if (t == 0) Wc2[g][0] = ptr;
        else {
          const int blk = (t + 1) / 2;        // t=1,2->b1; 3,4->b2; 5,6->b3
          if (t & 1) Wc1[g][blk] = ptr; else Wc2[g][blk] = ptr;
        }
      }
    }
  } else {
    // Single concatenated params buffer, jax pytree order:
    // 'groups' (sorted before 'stem'); block dicts sorted conv1, conv2, short.
    const float* P = (const float*)d_in[3];
    size_t off = 0;
    for (int g = 0; g < 3; ++g) {
      const int cin = CH[g], cout = CH[g + 1];
      for (int i = 0; i < 4; ++i) {
        const int ci = (i == 0) ? cin : cout;
        Wc1[g][i] = P + off; off += (size_t)cout * ci * 9;
        Wc2[g][i] = P + off; off += (size_t)cout * cout * 9;
        if (i == 0) { Wsh[g] = P + off; off += (size_t)cout * ci; }
      }
    }
    Wstem = P + off;
  }

  // Workspace: 12 slots of 131072 floats (covers the largest tensor 4*32*32*32).
  float* wsf = (float*)d_ws;
  const size_t SLOT = 131072;
  float* P0[3] = { wsf + 0 * SLOT, wsf + 1 * SLOT, wsf + 2 * SLOT };
  float* P1[3] = { wsf + 3 * SLOT, wsf + 4 * SLOT, wsf + 5 * SLOT };
  float* T[3]  = { wsf + 6 * SLOT, wsf + 7 * SLOT, wsf + 8 * SLOT };
  float* S[3]  = { wsf + 9 * SLOT, wsf + 10 * SLOT, wsf + 11 * SLOT };

  // Stem: 3 -> 16, k3 s1 p1, 32x32
  launch_ndconv(X, LO, UP, Wstem, P0[0], P0[1], P0[2],
                nullptr, nullptr, nullptr,
                3, 16, 3, 1, 1, 32, 32, 1.f, 0.f, stream);

  float* cur[3] = { P0[0], P0[1], P0[2] };
  float* alt[3] = { P1[0], P1[1], P1[2] };
  int curH = 32;
  static const int GS[3] = {1, 2, 2};

  for (int g = 0; g < 3; ++g) {
    const int cout = CH[g + 1];
    for (int i = 0; i < 4; ++i) {
      const int stride = (i == 0) ? GS[g] : 1;
      const int ci     = (i == 0) ? CH[g] : cout;
      const int Hout   = (curH + 2 - 3) / stride + 1;

      if (i == 0) {
        // 1x1 shortcut conv
        launch_ndconv(cur[0], cur[1], cur[2], Wsh[g], S[0], S[1], S[2],
                      nullptr, nullptr, nullptr,
                      ci, cout, 1, stride, 0, curH, curH, 1.f, 0.f, stream);
      }
      // conv1
      launch_ndconv(cur[0], cur[1], cur[2], Wc1[g][i], T[0], T[1], T[2],
                    nullptr, nullptr, nullptr,
                    ci, cout, 3, stride, 1, curH, curH, 1.f, 0.f, stream);
      // conv2 fused with 0.8/0.2 residual blend (shortcut conv or identity)
      const float* b0 = (i == 0) ? S[0] : cur[0];
      const float* b1 = (i == 0) ? S[1] : cur[1];
      const float* b2 = (i == 0) ? S[2] : cur[2];
      launch_ndconv(T[0], T[1], T[2], Wc2[g][i], alt[0], alt[1], alt[2],
                    b0, b1, b2,
                    cout, cout, 3, 1, 1, Hout, Hout, 0.8f, 0.2f, stream);

      for (int s2 = 0; s2 < 3; ++s2) { float* tp = cur[s2]; cur[s2] = alt[s2]; alt[s2] = tp; }
      curH = Hout;
    }
  }

  // Final GAP: (3, B, 128, 1, 1)
  const int BC = NB * 128;
  const int total = 3 * BC;
  gap_kernel<<<(total + 255) / 256, 256, 0, stream>>>(cur[0], cur[1], cur[2],
                                                      (float*)d_out, BC, curH * curH);
}